// TverskyProjectionLayer_74594991997509
// MI455X (gfx1250) — compile-verified
//
#include <hip/hip_runtime.h>
#include <hip/hip_bf16.h>
#include <math.h>

typedef float v2f __attribute__((ext_vector_type(2)));
typedef float v8f __attribute__((ext_vector_type(8)));

#define N_ROWS 2048
#define P_ROWS 512
#define D_DIM  1024
#define F_DIM  128
#define EPSV   1e-8f

__device__ __forceinline__ float softplus_f(float x) {
    return log1pf(expf(x));
}

// ---------------------------------------------------------------------------
// Normalize feature bank rows: fbn[r] = fb[r] / max(||fb[r]||, eps)
// one 256-thread block per row (D=1024 -> 4 elems/thread)
// ---------------------------------------------------------------------------
__global__ __launch_bounds__(256) void fbnorm_kernel(const float* __restrict__ fb,
                                                     float* __restrict__ fbn) {
    __shared__ float red[256];
    const int row = blockIdx.x;
    const int t   = threadIdx.x;
    const float* src = fb + (size_t)row * D_DIM;
    float s = 0.0f;
    for (int c = t; c < D_DIM; c += 256) { float v = src[c]; s += v * v; }
    red[t] = s;
    __syncthreads();
    for (int off = 128; off > 0; off >>= 1) {
        if (t < off) red[t] += red[t + off];
        __syncthreads();
    }
    const float scale = 1.0f / fmaxf(sqrtf(red[0]), EPSV);
    float* dst = fbn + (size_t)row * D_DIM;
    for (int c = t; c < D_DIM; c += 256) dst[c] = src[c] * scale;
}

// ---------------------------------------------------------------------------
// Per-row inverse L2 norm: invn[r] = 1 / max(||X[r]||, eps)
// ---------------------------------------------------------------------------
__global__ __launch_bounds__(256) void rowinvnorm_kernel(const float* __restrict__ X,
                                                         float* __restrict__ invn) {
    __shared__ float red[256];
    const int row = blockIdx.x;
    const int t   = threadIdx.x;
    const float* src = X + (size_t)row * D_DIM;
    float s = 0.0f;
    for (int c = t; c < D_DIM; c += 256) { float v = src[c]; s += v * v; }
    red[t] = s;
    __syncthreads();
    for (int off = 128; off > 0; off >>= 1) {
        if (t < off) red[t] += red[t + off];
        __syncthreads();
    }
    if (t == 0) invn[row] = 1.0f / fmaxf(sqrtf(red[0]), EPSV);
}

// ---------------------------------------------------------------------------
// out[row, col] = relu( invn[row] * (X[row,:] . fbn[col,:]) )
// fp32 WMMA 16x16x4: one wave computes a 16x16 tile, K loops over D in steps
// of 4 (256 v_wmma_f32_16x16x4_f32 per tile).
// A layout (16x4 f32): lanes 0-15 hold M=lane, K=(0,1); lanes 16-31 K=(2,3).
// B layout (4x16 f32): lanes 0-15 hold N=lane, K=(0,1); lanes 16-31 K=(2,3).
// C/D layout: VGPR v, lanes 0-15 -> M=v, N=lane; lanes 16-31 -> M=v+8.
// Block = 128 threads = 4 waves, one tile per wave; grids sized exactly so
// EXEC is all ones (WMMA requirement).
// ---------------------------------------------------------------------------
__global__ __launch_bounds__(128) void gemm_relu_wmma(const float* __restrict__ X,
                                                      const float* __restrict__ fbn,
                                                      const float* __restrict__ invn,
                                                      float* __restrict__ outp) {
    const int wid  = threadIdx.x >> 5;
    const int lane = threadIdx.x & 31;
    const int tile = blockIdx.x * 4 + wid;
    const int tf   = tile & 7;     // F_DIM/16 == 8 column tiles
    const int tm   = tile >> 3;
    const int half = lane >> 4;
    const int lm   = lane & 15;

    const float* arow = X   + (size_t)(tm * 16 + lm) * D_DIM + 2 * half;
    const float* brow = fbn + (size_t)(tf * 16 + lm) * D_DIM + 2 * half;

    v8f acc = {};
#pragma unroll 8
    for (int k = 0; k < D_DIM; k += 4) {
        v2f a2 = *(const v2f*)(arow + k);   // 8B aligned: k%4==0, offset 0 or 2
        v2f b2 = *(const v2f*)(brow + k);
        acc = __builtin_amdgcn_wmma_f32_16x16x4_f32(
            /*neg_a=*/false, a2, /*neg_b=*/false, b2,
            /*c_mod=*/(short)0, acc, /*reuse_a=*/false, /*reuse_b=*/false);
    }

    const int col = tf * 16 + lm;
#pragma unroll
    for (int v = 0; v < 8; ++v) {
        const int row = tm * 16 + half * 8 + v;
        const float val = acc[v] * invn[row];
        outp[(size_t)row * F_DIM + col] = fmaxf(val, 0.0f);
    }
}

// ---------------------------------------------------------------------------
// Pairwise Tversky reduction, algebraically reduced:
//   out[i,j] = (th+al+be)*S_min - al*S_a - be*S_b
//   S_min = sum_k min(ap,bp); S_a = sum_k ap*[bp>0]; S_b = sum_k [ap>0]*bp
// 16x16 output tile per 256-thread block; ap/bp tiles staged in LDS with
// row stride 132 floats (132%64==4 banks) to avoid stride-128 conflicts.
// ---------------------------------------------------------------------------
__global__ __launch_bounds__(256) void pairwise_kernel(const float* __restrict__ ap,
                                                       const float* __restrict__ bp,
                                                       const float* __restrict__ alpha_raw,
                                                       const float* __restrict__ beta_raw,
                                                       const float* __restrict__ theta_raw,
                                                       float* __restrict__ out) {
    __shared__ float apS[16][132];
    __shared__ float bpS[16][132];

    const int tx  = threadIdx.x;            // j within tile (0..15)
    const int ty  = threadIdx.y;            // i within tile (0..15)
    const int tid = ty * 16 + tx;
    const int j0  = blockIdx.x * 16;
    const int i0  = blockIdx.y * 16;

    // stage: 16 rows x 128 floats for each tile (512 float4 / 256 threads = 2)
    for (int p = tid; p < 16 * 32; p += 256) {
        const int r  = p >> 5;
        const int c4 = (p & 31) * 4;
        *(float4*)&apS[r][c4] = *(const float4*)(ap + (size_t)(i0 + r) * F_DIM + c4);
        *(float4*)&bpS[r][c4] = *(const float4*)(bp + (size_t)(j0 + r) * F_DIM + c4);
    }
    __syncthreads();

    float s_min = 0.0f, s_a = 0.0f, s_b = 0.0f;
#pragma unroll 4
    for (int k = 0; k < F_DIM; ++k) {
        const float av = apS[ty][k];
        const float bv = bpS[tx][k];
        s_min += fminf(av, bv);
        s_a   += (bv > 0.0f) ? av : 0.0f;
        s_b   += (av > 0.0f) ? bv : 0.0f;
    }

    const float al = softplus_f(alpha_raw[0]);
    const float be = softplus_f(beta_raw[0]);
    const float th = softplus_f(theta_raw[0]);
    out[(size_t)(i0 + ty) * P_ROWS + (j0 + tx)] =
        (th + al + be) * s_min - al * s_a - be * s_b;
}

// ---------------------------------------------------------------------------
extern "C" void kernel_launch(void* const* d_in, const int* in_sizes, int n_in,
                              void* d_out, int out_size, void* d_ws, size_t ws_size,
                              hipStream_t stream) {
    const float* input     = (const float*)d_in[0];   // 2048x1024
    const float* prototype = (const float*)d_in[1];   // 512x1024
    const float* fb        = (const float*)d_in[2];   // 128x1024
    const float* alpha_raw = (const float*)d_in[3];
    const float* beta_raw  = (const float*)d_in[4];
    const float* theta_raw = (const float*)d_in[5];
    float* out = (float*)d_out;                       // 2048x512

    float* w      = (float*)d_ws;
    float* fbn    = w;                                  // 128*1024
    float* inv_in = fbn + (size_t)F_DIM * D_DIM;        // 2048
    float* inv_pr = inv_in + N_ROWS;                    // 512
    float* ap     = inv_pr + P_ROWS;                    // 2048*128
    float* bp     = ap + (size_t)N_ROWS * F_DIM;        // 512*128
    // total ws use: ~1.76 MB

    fbnorm_kernel<<<F_DIM, 256, 0, stream>>>(fb, fbn);
    rowinvnorm_kernel<<<N_ROWS, 256, 0, stream>>>(input, inv_in);
    rowinvnorm_kernel<<<P_ROWS, 256, 0, stream>>>(prototype, inv_pr);

    // tiles = (rows/16)*(128/16); 4 tiles (waves) per block
    gemm_relu_wmma<<<(N_ROWS / 16) * (F_DIM / 16) / 4, 128, 0, stream>>>(input, fbn, inv_in, ap);
    gemm_relu_wmma<<<(P_ROWS / 16) * (F_DIM / 16) / 4, 128, 0, stream>>>(prototype, fbn, inv_pr, bp);

    dim3 grid(P_ROWS / 16, N_ROWS / 16);
    pairwise_kernel<<<grid, dim3(16, 16), 0, stream>>>(ap, bp, alpha_raw, beta_raw, theta_raw, out);
}